// SelectiveScanPyTorch_17446157156533
// MI455X (gfx1250) — compile-verified
//
#include <hip/hip_runtime.h>
#include <hip/hip_bf16.h>
#include <cmath>

// ---------------------------------------------------------------------------
// Mamba selective scan, CDNA5 (gfx1250) chunked-WMMA formulation, round 2.
//
// One wave32 per (b,d) channel; chunk CH=16 timesteps on lanes 0-15 with
// lanes 16-31 replicating (WMMA operand halves become v_cndmask selects).
//
//   Ct[n,l] = C[n,l] * exp((S_l - Sc) A_n)
//   Bt[n,i] = B[n,i] * dt_i*u_i * exp((Sc - S_i) A_n)
//   G = Ct^T @ Bt              -> 4x v_wmma_f32_16x16x4_f32 (K=16)
//   y_l = tril-rowsum(G) + sum_n Ct[n,l]*dC_n*x_n,   dC_n = exp(Sc A_n)
//   x_n <- dC_n*(dC_n*x_n + sum_i Bt[n,i])
//
// Round-2 change: ALL cross-lane traffic moved off LDS (ds_bpermute) onto
// DPP16 (row_shr scan, quad_perm/mirror butterflies) and v_permlanex16
// (half-wave swap).  The only memory ops left are the coalesced input
// streams (~67MB total => ~3us HBM floor) plus global_prefetch_b8 ahead of
// the scan.  Trans work (exp/rcp) co-executes with VALU; matmul rides WMMA.
// ---------------------------------------------------------------------------

#define BSZ 2
#define DM  2048
#define LEN 1024
#define NST 16
#define CH  16
#define NCHUNK (LEN / CH)

typedef float v2f __attribute__((ext_vector_type(2)));
typedef float v8f __attribute__((ext_vector_type(8)));

// DPP16 control codes
#define DPP_QUAD_XOR1   0xB1   // quad_perm [1,0,3,2]
#define DPP_QUAD_XOR2   0x4E   // quad_perm [2,3,0,1]
#define DPP_ROW_SHR(n)  (0x110 | (n))
#define DPP_HALF_MIRROR 0x141  // lane ^ 7 within groups of 8
#define DPP_ROW_MIRROR  0x140  // lane ^ 15 within rows of 16

template <int CTRL>
__device__ __forceinline__ float dpp0(float v) {
  // update_dpp with old=0, bound_ctrl=1: out-of-bounds lanes read 0.
  int r = __builtin_amdgcn_update_dpp(0, __float_as_int(v), CTRL, 0xF, 0xF, true);
  return __int_as_float(r);
}

// Sum across each 16-lane row (result replicated to all 16 lanes of the row).
__device__ __forceinline__ float row_sum16(float v) {
  v += dpp0<DPP_QUAD_XOR1>(v);    // pairs
  v += dpp0<DPP_QUAD_XOR2>(v);    // quads
  v += dpp0<DPP_HALF_MIRROR>(v);  // groups of 8 (quads uniform -> xor7 ok)
  v += dpp0<DPP_ROW_MIRROR>(v);   // row of 16  (octets uniform -> xor15 ok)
  return v;
}

// Inclusive prefix sum across each 16-lane row.
__device__ __forceinline__ float row_scan16(float v) {
  v += dpp0<DPP_ROW_SHR(1)>(v);
  v += dpp0<DPP_ROW_SHR(2)>(v);
  v += dpp0<DPP_ROW_SHR(4)>(v);
  v += dpp0<DPP_ROW_SHR(8)>(v);
  return v;
}

// Swap value with lane^16 (other half-wave), pure VALU via v_permlanex16.
__device__ __forceinline__ float half_swap(float v) {
#if __has_builtin(__builtin_amdgcn_permlanex16)
  int r = __builtin_amdgcn_permlanex16(__float_as_int(v), __float_as_int(v),
                                       0x76543210, 0xFEDCBA98, false, false);
  return __int_as_float(r);
#else
  return __shfl(v, ((int)(threadIdx.x & 31)) ^ 16, 32);
#endif
}

__global__ __launch_bounds__(256) void mamba_scan_wmma(
    const float* __restrict__ u,  const float* __restrict__ delta,
    const float* __restrict__ A,  const float* __restrict__ Bm,
    const float* __restrict__ Cm, const float* __restrict__ Dv,
    const float* __restrict__ z,  const float* __restrict__ dbias,
    const int*  __restrict__ softplus_flag,
    float* __restrict__ y)
{
  const int tid  = blockIdx.x * blockDim.x + threadIdx.x;
  const int wave = tid >> 5;
  const int lane = threadIdx.x & 31;
  const int la   = lane & 15;          // replicated half-wave index
  if (wave >= BSZ * DM) return;        // wave-uniform guard

  const int b = wave / DM;
  const int d = wave % DM;

  const size_t chbase = ((size_t)b * DM + d) * LEN;
  const float* up  = u + chbase;
  const float* dtp = delta + chbase;
  const float* zp  = z + chbase;
  float*       yp  = y + chbase;
  const float* Bp = Bm + (size_t)b * NST * LEN;
  const float* Cp = Cm + (size_t)b * NST * LEN;

  float An[NST];
#pragma unroll
  for (int n = 0; n < NST; ++n) An[n] = A[d * NST + n];
  const float db = dbias[d];
  const float Dd = Dv[d];
  const int   sp = softplus_flag[0];

  float x[NST];
#pragma unroll
  for (int n = 0; n < NST; ++n) x[n] = 0.f;

  for (int c = 0; c < NCHUNK; ++c) {
    const int idx = c * CH + la;       // lanes 16-31 replicate lanes 0-15

    // Prefetch next chunk's elementwise streams into cache (speculative).
    if (c + 1 < NCHUNK) {
      __builtin_prefetch(dtp + idx + CH, 0, 1);
      __builtin_prefetch(up  + idx + CH, 0, 1);
      __builtin_prefetch(zp  + idx + CH, 0, 1);
    }

    float dtr = dtp[idx] + db;
    float dtv = sp ? ((dtr > 20.f) ? dtr : log1pf(__expf(dtr))) : dtr;
    const float uu = up[idx];
    const float zz = zp[idx];

    // Segmented (per-16-lane) scan + total via DPP, no LDS.
    const float S    = row_scan16(dtv);       // inclusive cumsum of dt
    const float T    = row_sum16(dtv);        // chunk total
    const float Sc   = 0.5f * T;              // midpoint centering
    const float Srel = S - Sc;
    const float dtu  = dtv * uu;

    float ct[NST], bt[NST], dC[NST];
    float ycar = 0.f;
#pragma unroll
    for (int n = 0; n < NST; ++n) {
      const float aS   = An[n];
      const float e    = __expf(Srel * aS);          // exp((S_l-Sc)A_n)
      const float einv = __builtin_amdgcn_rcpf(e);   // exp((Sc-S_l)A_n)
      const float dc   = __expf(Sc * aS);            // exp(Sc A_n) <= 1
      ct[n] = Cp[n * LEN + idx] * e;
      bt[n] = Bp[n * LEN + idx] * dtu * einv;
      dC[n] = dc;
      ycar += ct[n] * (dc * x[n]);                   // carry-in term
    }

    // ---------------- G = Ct^T (16l x 16n) @ Bt (16n x 16i) via WMMA ------
    v8f G = {0.f, 0.f, 0.f, 0.f, 0.f, 0.f, 0.f, 0.f};
#pragma unroll
    for (int k0 = 0; k0 < NST; k0 += 4) {
      v2f a, bb;
      a.x  = (lane < 16) ? ct[k0]     : ct[k0 + 2];
      a.y  = (lane < 16) ? ct[k0 + 1] : ct[k0 + 3];
      bb.x = (lane < 16) ? bt[k0]     : bt[k0 + 2];
      bb.y = (lane < 16) ? bt[k0 + 1] : bt[k0 + 3];
      G = __builtin_amdgcn_wmma_f32_16x16x4_f32(false, a, false, bb,
                                                (short)0, G, false, false);
    }

    // tril row-sums of the D tile: VGPR r = rows r (lanes0-15) / r+8
    // (lanes16-31); lanes index columns i.  Mask i<=l, DPP butterfly,
    // cross-half fetch via v_permlanex16.
    float ysum = 0.f;
#pragma unroll
    for (int r = 0; r < 8; ++r) {
      const int row = r + ((lane & 16) ? 8 : 0);
      float g = (la <= row) ? G[r] : 0.f;
      g = row_sum16(g);
      const float gOpp = half_swap(g);     // other half's row-sum
      if ((la & 7) == r) ysum = (la < 8) ? g : gOpp;
    }

    // State update: x_n <- dC*(dC*x_n + sum_i bt[n,i])  (DPP reduce)
#pragma unroll
    for (int n = 0; n < NST; ++n) {
      const float s = row_sum16(bt[n]);
      x[n] = dC[n] * (dC[n] * x[n] + s);
    }

    // Skip connection + SiLU(z) gate; lanes 0-15 hold the valid rows.
    const float yl  = ysum + ycar;
    const float sig = __builtin_amdgcn_rcpf(1.f + __expf(-zz));
    const float out = (yl + uu * Dd) * (zz * sig);
    if (lane < 16) yp[idx] = out;
  }
}

extern "C" void kernel_launch(void* const* d_in, const int* in_sizes, int n_in,
                              void* d_out, int out_size, void* d_ws, size_t ws_size,
                              hipStream_t stream) {
  (void)in_sizes; (void)n_in; (void)out_size; (void)d_ws; (void)ws_size;
  const float* u     = (const float*)d_in[0];
  const float* delta = (const float*)d_in[1];
  const float* A     = (const float*)d_in[2];
  const float* B     = (const float*)d_in[3];
  const float* C     = (const float*)d_in[4];
  const float* D     = (const float*)d_in[5];
  const float* z     = (const float*)d_in[6];
  const float* dbias = (const float*)d_in[7];
  const int*   spf   = (const int*)d_in[8];
  float*       y     = (float*)d_out;

  const int waves   = BSZ * DM;                // 4096 waves, one per channel
  const int threads = 256;                     // 8 waves per block
  const int blocks  = (waves * 32) / threads;  // 512 blocks
  mamba_scan_wmma<<<blocks, threads, 0, stream>>>(u, delta, A, B, C, D, z,
                                                  dbias, spf, y);
}